// TreeLSTMModel_3942779978310
// MI455X (gfx1250) — compile-verified
//
#include <hip/hip_runtime.h>
#include <hip/hip_bf16.h>

// ---------------------------------------------------------------------------
// Forest TreeLSTM for MI455X (gfx1250, wave32).  All GEMMs use
// V_WMMA_F32_16X16X4_F32 (native fp32 WMMA): the problem is HBM-bandwidth
// bound (~1.3 GB traffic vs ~28 GFLOP fp32), so fp32 WMMA is the right
// precision -- no benefit (and accuracy risk) from f16/fp8 downcasts.
// ---------------------------------------------------------------------------

typedef __attribute__((ext_vector_type(2))) float v2f;
typedef __attribute__((ext_vector_type(8))) float v8f;

#define N_TREES   512
#define NPT       1023          // nodes per tree
#define N_NODES   (N_TREES * NPT)
#define HID       64

__device__ __forceinline__ float sigf(float x) {
    return 1.0f / (1.0f + __expf(-x));
}
__device__ __forceinline__ float tanhf_fast(float x) {
    float e = __expf(-2.0f * x);
    return (1.0f - e) / (1.0f + e);
}
__device__ __forceinline__ v8f wmma_f32(v2f a, v2f b, v8f c) {
    // 8-arg form: (neg_a, A, neg_b, B, c_mod, C, reuse_a, reuse_b)
    return __builtin_amdgcn_wmma_f32_16x16x4_f32(false, a, false, b,
                                                 (short)0, c, false, false);
}

// ---------------------------------------------------------------------------
// Kernel 1: xw[N][256] = features[N][64] @ [W_iou; W_f]^T + [b_iou; b_f]
// cols 0..191 -> iou_x, cols 192..255 -> f_x.
// One wave owns a 16-row tile; 8 waves/block -> 128 rows/block; 4092 blocks.
// ---------------------------------------------------------------------------
__global__ void __launch_bounds__(256)
gemm_xw_kernel(const float* __restrict__ feat,
               const float* __restrict__ w_iou, const float* __restrict__ b_iou,
               const float* __restrict__ w_f,   const float* __restrict__ b_f,
               float* __restrict__ xw)
{
    const int wave   = threadIdx.x >> 5;
    const int lane   = threadIdx.x & 31;
    const int lane16 = lane & 15;
    const int half   = lane >> 4;            // 0: K={0,1}, 1: K={2,3}
    const int row0   = blockIdx.x * 128 + wave * 16;
    const int koff   = half * 2;

    __builtin_prefetch(w_iou + (threadIdx.x << 4), 0, 0);
    __builtin_prefetch(w_f   + (threadIdx.x << 3), 0, 0);

    // A fragments: this lane's row, half of K (16 k-steps of 4)
    const int mA = row0 + lane16;
    v2f aF[16];
#pragma unroll
    for (int kk = 0; kk < 16; ++kk)
        aF[kk] = *(const v2f*)(feat + mA * 64 + 4 * kk + koff);

    for (int ct = 0; ct < 16; ++ct) {
        const int col = ct * 16 + lane16;                 // 0..255
        const float* wsrc = (col < 192) ? (w_iou + col * 64)
                                        : (w_f + (col - 192) * 64);
        const float* bsrc = (col < 192) ? (b_iou + col) : (b_f + (col - 192));
        const float bias = *bsrc;

        v8f acc;
#pragma unroll
        for (int r = 0; r < 8; ++r) acc[r] = bias;

#pragma unroll
        for (int kk = 0; kk < 16; ++kk) {
            v2f b = *(const v2f*)(wsrc + 4 * kk + koff);
            acc = wmma_f32(aF[kk], b, acc);
        }
        // C/D layout: vgpr r, lanes 0-15 -> M=r, lanes 16-31 -> M=8+r
#pragma unroll
        for (int r = 0; r < 8; ++r)
            xw[(row0 + half * 8 + r) * 256 + col] = acc[r];
    }
}

// ---------------------------------------------------------------------------
// Kernel 2: leaves (heap idx 511..1022 in each tree): pure elementwise gates.
// ---------------------------------------------------------------------------
__global__ void __launch_bounds__(256)
leaf_kernel(const float* __restrict__ xw,
            float* __restrict__ h, float* __restrict__ c)
{
    const int idx  = blockIdx.x * blockDim.x + threadIdx.x;   // 512*512*64
    const int col  = idx & 63;
    const int li   = (idx >> 6) & 511;
    const int tree = idx >> 15;
    const int node = tree * NPT + 511 + li;
    const int base = node * 256;

    float i = sigf(xw[base + col]);
    float o = sigf(xw[base + 64 + col]);
    float u = tanhf_fast(xw[base + 128 + col]);
    float cv = i * u;
    c[node * 64 + col] = cv;
    h[node * 64 + col] = o * tanhf_fast(cv);
}

// ---------------------------------------------------------------------------
// Kernel 3: one tree level (order n >= 1).  Per parent p with children l,r:
//   iou = iou_x[p] + (h[l]+h[r]) @ U_iou^T
//   f_l = sig(f_x[p] + h[l] @ U_f^T),  f_r likewise (A_r = A_sum - A_l)
//   c[p] = i*u + f_l*c[l] + f_r*c[r] ; h[p] = o*tanh(c[p])
// One wave = 16 parents; block = 128 parents (level sizes are multiples).
// ---------------------------------------------------------------------------
__global__ void __launch_bounds__(256)
level_kernel(const float* __restrict__ xw,
             const float* __restrict__ u_iou, const float* __restrict__ u_f,
             float* __restrict__ h, float* __restrict__ c,
             int sOff, int cptShift)
{
    const int wave   = threadIdx.x >> 5;
    const int lane   = threadIdx.x & 31;
    const int lane16 = lane & 15;
    const int half   = lane >> 4;
    const int koff   = half * 2;
    const int p0     = blockIdx.x * 128 + wave * 16;
    const int cptM1  = (1 << cptShift) - 1;

    __builtin_prefetch(u_iou + (threadIdx.x << 4), 0, 0);
    __builtin_prefetch(u_f   + (threadIdx.x << 3), 0, 0);

    // node ids for this lane's A-row (M = lane16)
    {
    }
    const int lpA   = p0 + lane16;
    const int heapA = sOff + (lpA & cptM1);
    const int nodeA = (lpA >> cptShift) * NPT + heapA;
    const int clA   = nodeA - heapA + 2 * heapA + 1;   // tree*NPT + 2*heap+1
    const int crA   = clA + 1;
    (void)nodeA;

    // node ids for this lane's C-rows (M = half*8 + r)
    int nodeR[8], clR[8], crR[8];
#pragma unroll
    for (int r = 0; r < 8; ++r) {
        int lp   = p0 + half * 8 + r;
        int heap = sOff + (lp & cptM1);
        int nd   = (lp >> cptShift) * NPT + heap;
        nodeR[r] = nd;
        clR[r]   = nd - heap + 2 * heap + 1;
        crR[r]   = clR[r] + 1;
    }

    // A fragments: h_sum and h_left (h_right = sum - left)
    v2f aS[16], aL[16];
#pragma unroll
    for (int kk = 0; kk < 16; ++kk) {
        int kb = 4 * kk + koff;
        v2f hl = *(const v2f*)(h + clA * 64 + kb);
        v2f hr = *(const v2f*)(h + crA * 64 + kb);
        aL[kk] = hl;
        aS[kk] = hl + hr;
    }

    for (int ct = 0; ct < 4; ++ct) {
        const int col = ct * 16 + lane16;   // 0..63 within each gate block

        v8f accI, accO, accU, accFl, accFr;
#pragma unroll
        for (int r = 0; r < 8; ++r) {
            const int nb = nodeR[r] * 256;
            accI[r]  = xw[nb + col];
            accO[r]  = xw[nb + 64 + col];
            accU[r]  = xw[nb + 128 + col];
            float fx = xw[nb + 192 + col];
            accFl[r] = fx;
            accFr[r] = fx;
        }

#pragma unroll
        for (int kk = 0; kk < 16; ++kk) {
            const int kb = 4 * kk + koff;
            v2f bI = *(const v2f*)(u_iou + col * 64 + kb);
            v2f bO = *(const v2f*)(u_iou + (64 + col) * 64 + kb);
            v2f bU = *(const v2f*)(u_iou + (128 + col) * 64 + kb);
            v2f bF = *(const v2f*)(u_f + col * 64 + kb);
            v2f aR = aS[kk] - aL[kk];
            accI  = wmma_f32(aS[kk], bI, accI);
            accO  = wmma_f32(aS[kk], bO, accO);
            accU  = wmma_f32(aS[kk], bU, accU);
            accFl = wmma_f32(aL[kk], bF, accFl);
            accFr = wmma_f32(aR,     bF, accFr);
        }

#pragma unroll
        for (int r = 0; r < 8; ++r) {
            float i  = sigf(accI[r]);
            float o  = sigf(accO[r]);
            float u  = tanhf_fast(accU[r]);
            float fl = sigf(accFl[r]);
            float fr = sigf(accFr[r]);
            float cl = c[clR[r] * 64 + col];
            float cr = c[crR[r] * 64 + col];
            float cn = i * u + fl * cl + fr * cr;
            c[nodeR[r] * 64 + col] = cn;
            h[nodeR[r] * 64 + col] = o * tanhf_fast(cn);
        }
    }
}

// ---------------------------------------------------------------------------
// Kernel 4: per-tree mean of h over 1023 nodes, then relu -> xmean[512][64]
// ---------------------------------------------------------------------------
__global__ void __launch_bounds__(64)
tree_mean_kernel(const float* __restrict__ h, float* __restrict__ xmean)
{
    const int tree = blockIdx.x;
    const int col  = threadIdx.x;
    const float* p = h + tree * NPT * 64 + col;
    float s = 0.0f;
    for (int ni = 0; ni < NPT; ++ni) s += p[ni * 64];
    s *= (1.0f / (float)NPT);
    xmean[tree * 64 + col] = fmaxf(s, 0.0f);
}

// ---------------------------------------------------------------------------
// Kernel 5: MLP head with WMMA.  512 trees, 128/block, 16/wave.
// C-layout -> A-layout re-staging between layers goes through LDS.
// ---------------------------------------------------------------------------
__global__ void __launch_bounds__(256)
mlp_kernel(const float* __restrict__ xmean,
           const float* __restrict__ lin0_w, const float* __restrict__ lin0_b,
           const float* __restrict__ lin1_w, const float* __restrict__ lin1_b,
           const float* __restrict__ out_w,  const float* __restrict__ out_b,
           float* __restrict__ out)
{
    __shared__ float xs[8 * 16 * 64];   // 32 KB: per-wave 16x64 tile

    const int wave   = threadIdx.x >> 5;
    const int lane   = threadIdx.x & 31;
    const int lane16 = lane & 15;
    const int half   = lane >> 4;
    const int koff   = half * 2;
    const int row0   = blockIdx.x * 128 + wave * 16;   // tree base
    float* myx = xs + wave * (16 * 64);

    // ---- layer 0: A from global xmean ----
    v2f a0[16];
#pragma unroll
    for (int kk = 0; kk < 16; ++kk)
        a0[kk] = *(const v2f*)(xmean + (row0 + lane16) * 64 + 4 * kk + koff);

    for (int ct = 0; ct < 4; ++ct) {
        const int col = ct * 16 + lane16;
        v8f acc;
        float bias = lin0_b[col];
#pragma unroll
        for (int r = 0; r < 8; ++r) acc[r] = bias;
#pragma unroll
        for (int kk = 0; kk < 16; ++kk) {
            v2f b = *(const v2f*)(lin0_w + col * 64 + 4 * kk + koff);
            acc = wmma_f32(a0[kk], b, acc);
        }
#pragma unroll
        for (int r = 0; r < 8; ++r)
            myx[(half * 8 + r) * 64 + col] = fmaxf(acc[r], 0.0f);
    }
    __syncthreads();

    // ---- layer 1: A from LDS ----
    v2f a1[16];
#pragma unroll
    for (int kk = 0; kk < 16; ++kk) {
        int kb = 4 * kk + koff;
        a1[kk] = (v2f){myx[lane16 * 64 + kb], myx[lane16 * 64 + kb + 1]};
    }
    __syncthreads();

    for (int ct = 0; ct < 4; ++ct) {
        const int col = ct * 16 + lane16;
        v8f acc;
        float bias = lin1_b[col];
#pragma unroll
        for (int r = 0; r < 8; ++r) acc[r] = bias;
#pragma unroll
        for (int kk = 0; kk < 16; ++kk) {
            v2f b = *(const v2f*)(lin1_w + col * 64 + 4 * kk + koff);
            acc = wmma_f32(a1[kk], b, acc);
        }
#pragma unroll
        for (int r = 0; r < 8; ++r)
            myx[(half * 8 + r) * 64 + col] = fmaxf(acc[r], 0.0f);
    }
    __syncthreads();

    // ---- output layer: y = x2 . out_w + out_b ----
    const int cstart = half * 32;
    float partial = 0.0f;
#pragma unroll
    for (int j = 0; j < 32; ++j)
        partial += myx[lane16 * 64 + cstart + j] * out_w[cstart + j];
    partial += __shfl_xor(partial, 16, 32);    // combine the two K-halves
    if (half == 0)
        out[row0 + lane16] = partial + out_b[0];
}

// ---------------------------------------------------------------------------
extern "C" void kernel_launch(void* const* d_in, const int* in_sizes, int n_in,
                              void* d_out, int out_size, void* d_ws, size_t ws_size,
                              hipStream_t stream) {
    const float* feat   = (const float*)d_in[0];
    const float* w_iou  = (const float*)d_in[4];
    const float* b_iou  = (const float*)d_in[5];
    const float* u_iou  = (const float*)d_in[6];
    const float* w_f    = (const float*)d_in[7];
    const float* b_f    = (const float*)d_in[8];
    const float* u_f    = (const float*)d_in[9];
    const float* lin0_w = (const float*)d_in[10];
    const float* lin0_b = (const float*)d_in[11];
    const float* lin1_w = (const float*)d_in[12];
    const float* lin1_b = (const float*)d_in[13];
    const float* out_w  = (const float*)d_in[14];
    const float* out_b  = (const float*)d_in[15];
    float* out = (float*)d_out;

    float* xw    = (float*)d_ws;                       // N x 256
    float* h     = xw + (size_t)N_NODES * 256;         // N x 64
    float* c     = h + (size_t)N_NODES * 64;           // N x 64
    float* xmean = c + (size_t)N_NODES * 64;           // 512 x 64

    // 1) fused input projections (N x 64) @ (64 x 256)
    gemm_xw_kernel<<<N_NODES / 128, 256, 0, stream>>>(feat, w_iou, b_iou,
                                                      w_f, b_f, xw);
    // 2) leaves
    leaf_kernel<<<(N_TREES * 512 * 64) / 256, 256, 0, stream>>>(xw, h, c);
    // 3) internal levels, leaves -> root
    for (int n = 1; n <= 9; ++n) {
        int sh = 9 - n;                    // log2(parents per tree)
        int sOff = (1 << sh) - 1;          // heap offset of first parent
        int totalParents = N_TREES << sh;  // multiple of 128 for all levels
        level_kernel<<<totalParents / 128, 256, 0, stream>>>(
            xw, u_iou, u_f, h, c, sOff, sh);
    }
    // 4) per-tree mean + relu
    tree_mean_kernel<<<N_TREES, 64, 0, stream>>>(h, xmean);
    // 5) MLP head
    mlp_kernel<<<N_TREES / 128, 256, 0, stream>>>(xmean, lin0_w, lin0_b,
                                                  lin1_w, lin1_b,
                                                  out_w, out_b, out);
}